// MoEFeedForward_31894427140183
// MI455X (gfx1250) — compile-verified
//
#include <hip/hip_runtime.h>
#include <hip/hip_bf16.h>

// Problem constants (from reference setup_inputs)
#define Bq 2048
#define Hq 512
#define Iq 1024
#define Eq 8

typedef __attribute__((ext_vector_type(2))) float v2f;
typedef __attribute__((ext_vector_type(8))) float v8f;

// LDS pitches (floats). Even (8B-aligned v2f loads), padded vs 64-bank LDS.
#define XS_PITCH  520   // 16 rows x 512 cols of gathered x
#define ACT_PITCH 1032  // 16 rows x 1024 cols of SwiGLU activations

// ---------------------------------------------------------------------------
// Kernel 0: zero d_out and routing workspace; pre-fill expert lists with
// (token 0, weight 0) padding so partial tiles need no masking.
// ---------------------------------------------------------------------------
__global__ __launch_bounds__(256) void moe_init_kernel(float* __restrict__ out,
                                                       int* __restrict__ counts,
                                                       int* __restrict__ tok,
                                                       float* __restrict__ wt) {
  const int i = blockIdx.x * 256 + threadIdx.x;
  if (i < Bq * Hq) out[i] = 0.0f;
  if (i < Eq) counts[i] = 0;
  if (i < Eq * Bq) { tok[i] = 0; wt[i] = 0.0f; }
}

// ---------------------------------------------------------------------------
// Kernel 1: router. logits = x@Wr + br + gumbel, top-2, softmax over the two
// kept logits, scatter (token, weight) into per-expert lists via atomic slots.
// ---------------------------------------------------------------------------
__global__ __launch_bounds__(256) void moe_router_kernel(
    const float* __restrict__ x, const float* __restrict__ Wr,
    const float* __restrict__ br, const float* __restrict__ gum,
    int* __restrict__ counts, int* __restrict__ tok, float* __restrict__ wt) {
  const int b = blockIdx.x * 256 + threadIdx.x;
  if (b >= Bq) return;

  float acc[Eq];
#pragma unroll
  for (int e = 0; e < Eq; ++e) acc[e] = 0.0f;

  const float* xr = x + (size_t)b * Hq;
  for (int h = 0; h < Hq; ++h) {
    const float xv = xr[h];
    const float* wr = Wr + (size_t)h * Eq;  // uniform across lanes -> broadcast
#pragma unroll
    for (int e = 0; e < Eq; ++e) acc[e] += xv * wr[e];
  }
#pragma unroll
  for (int e = 0; e < Eq; ++e) acc[e] += br[e] + gum[(size_t)b * Eq + e];

  // top-2 (first occurrence wins ties, matching lax.top_k)
  float v0 = -3.0e38f; int i0 = 0;
#pragma unroll
  for (int e = 0; e < Eq; ++e) { if (acc[e] > v0) { v0 = acc[e]; i0 = e; } }
  float v1 = -3.0e38f; int i1 = 0;
#pragma unroll
  for (int e = 0; e < Eq; ++e) { if (e != i0 && acc[e] > v1) { v1 = acc[e]; i1 = e; } }

  // softmax over [v0, v1] with v0 >= v1 (exact path: arg <= 0, expf precise)
  const float w0 = 1.0f / (1.0f + __expf(v1 - v0));
  const float w1 = 1.0f - w0;

  int s0 = atomicAdd(&counts[i0], 1);
  tok[i0 * Bq + s0] = b; wt[i0 * Bq + s0] = w0;
  int s1 = atomicAdd(&counts[i1], 1);
  tok[i1 * Bq + s1] = b; wt[i1 * Bq + s1] = w1;
}

// ---------------------------------------------------------------------------
// Kernel 2: grouped expert FFN over 16-token tiles using fp32 WMMA.
//   GEMM1: X(16x512) @ W_in[e](512x2048) -> h, fused bias + SwiGLU -> act
//   GEMM2: act(16x1024) @ W_out[e](1024x512), bias, row-scale by softmax
//          weight, atomic combine into out.
// 256 threads = 8 waves. Dynamic LDS: xs[16][520] + act[16][1032].
// Four independent WMMA accumulator chains share each A-fragment load.
// ---------------------------------------------------------------------------
__global__ __launch_bounds__(256) void moe_ffn_kernel(
    const float* __restrict__ x, const float* __restrict__ Win,
    const float* __restrict__ bin, const float* __restrict__ Wout,
    const float* __restrict__ bout, const int* __restrict__ counts,
    const int* __restrict__ tok, const float* __restrict__ wt,
    float* __restrict__ out) {
  const int e    = blockIdx.x >> 7;    // / 128 tiles per expert
  const int tile = blockIdx.x & 127;
  const int cnt  = counts[e];
  if (tile * 16 >= cnt) return;        // uniform per block

  extern __shared__ float smem[];
  float* xs  = smem;                   // [16][XS_PITCH]
  float* act = smem + 16 * XS_PITCH;   // [16][ACT_PITCH]
  __shared__ int   tokS[16];
  __shared__ float wS[16];

  const int tid  = threadIdx.x;
  const int lane = tid & 31;
  const int wave = tid >> 5;
  const int hi   = lane >> 4;          // half-wave (K split for A/B, M split for C)
  const int lm   = lane & 15;          // N (or M) index within 16

  // ---- Phase A: tile metadata + gather 16 rows of x into LDS ----
  if (tid < 16) {
    tokS[tid] = tok[e * Bq + tile * 16 + tid];
    wS[tid]   = wt [e * Bq + tile * 16 + tid];   // padded rows: weight 0
  }
  __syncthreads();
  for (int i = tid; i < 16 * (Hq / 4); i += 256) {
    const int r = i >> 7, c4 = i & 127;
    const float4 v = ((const float4*)(x + (size_t)tokS[r] * Hq))[c4];
    float* dst = xs + r * XS_PITCH + c4 * 4;
    dst[0] = v.x; dst[1] = v.y; dst[2] = v.z; dst[3] = v.w;
  }
  __syncthreads();

  // ---- Phase B: GEMM1 (K=512) + bias + SwiGLU -> act in LDS ----
  // Wave w owns x1 cols [w*128, w*128+128) and the paired x2 cols (+1024).
  // Process two 16-col subtiles at once: 4 accumulator chains per A load.
  const float* We = Win + (size_t)e * Hq * (2 * Iq);
  for (int s = 0; s < 4; ++s) {
    const int n = wave * 128 + s * 32;           // cols [n, n+32)
    v8f c1a = {}, c1b = {}, c2a = {}, c2b = {};
#pragma unroll 2
    for (int k0 = 0; k0 < Hq; k0 += 4) {
      // A 16x4 fp32: lane's two elements are K-consecutive -> one b64 LDS load
      v2f a = *(const v2f*)(xs + lm * XS_PITCH + k0 + 2 * hi);
      const float* bp = We + (size_t)(k0 + 2 * hi) * (2 * Iq) + n + lm;
      v2f b1a, b1b, b2a, b2b;
      b1a.x = bp[0];        b1a.y = bp[2 * Iq];
      b1b.x = bp[16];       b1b.y = bp[2 * Iq + 16];
      b2a.x = bp[Iq];       b2a.y = bp[2 * Iq + Iq];
      b2b.x = bp[Iq + 16];  b2b.y = bp[2 * Iq + Iq + 16];
      c1a = __builtin_amdgcn_wmma_f32_16x16x4_f32(false, a, false, b1a,
                                                  (short)0, c1a, false, false);
      c1b = __builtin_amdgcn_wmma_f32_16x16x4_f32(false, a, false, b1b,
                                                  (short)0, c1b, false, false);
      c2a = __builtin_amdgcn_wmma_f32_16x16x4_f32(false, a, false, b2a,
                                                  (short)0, c2a, false, false);
      c2b = __builtin_amdgcn_wmma_f32_16x16x4_f32(false, a, false, b2b,
                                                  (short)0, c2b, false, false);
    }
    // Fused bias + SwiGLU epilogue (fast rcp: ~1 ulp on sigmoid)
    auto swiglu_store = [&](const v8f& c1, const v8f& c2, int ncol) {
      const float bb1 = bin[(size_t)e * (2 * Iq) + ncol + lm];
      const float bb2 = bin[(size_t)e * (2 * Iq) + Iq + ncol + lm];
#pragma unroll
      for (int r = 0; r < 8; ++r) {
        // C layout: VGPR r -> row M = r + 8*hi, col N = lm
        const float u  = c1[r] + bb1;
        const float g  = c2[r] + bb2;
        const float sg = __fdividef(1.0f, 1.0f + __expf(-g));
        act[(r + 8 * hi) * ACT_PITCH + ncol + lm] = u * sg;
      }
    };
    swiglu_store(c1a, c2a, n);
    swiglu_store(c1b, c2b, n + 16);
  }
  __syncthreads();

  // ---- Phase C: GEMM2 (K=1024), bias, weight-scale, atomic combine ----
  // Wave w owns out cols [w*64, w*64+64): 4 accumulator chains per A load.
  const float* Wo = Wout + (size_t)e * Iq * Hq;
  {
    const int n = wave * 64;
    v8f c0 = {}, c1 = {}, c2 = {}, c3 = {};
#pragma unroll 2
    for (int k0 = 0; k0 < Iq; k0 += 4) {
      v2f a = *(const v2f*)(act + lm * ACT_PITCH + k0 + 2 * hi);
      const float* bp = Wo + (size_t)(k0 + 2 * hi) * Hq + n + lm;
      v2f b0, b1, b2, b3;
      b0.x = bp[0];   b0.y = bp[Hq];
      b1.x = bp[16];  b1.y = bp[Hq + 16];
      b2.x = bp[32];  b2.y = bp[Hq + 32];
      b3.x = bp[48];  b3.y = bp[Hq + 48];
      c0 = __builtin_amdgcn_wmma_f32_16x16x4_f32(false, a, false, b0,
                                                 (short)0, c0, false, false);
      c1 = __builtin_amdgcn_wmma_f32_16x16x4_f32(false, a, false, b1,
                                                 (short)0, c1, false, false);
      c2 = __builtin_amdgcn_wmma_f32_16x16x4_f32(false, a, false, b2,
                                                 (short)0, c2, false, false);
      c3 = __builtin_amdgcn_wmma_f32_16x16x4_f32(false, a, false, b3,
                                                 (short)0, c3, false, false);
    }
    auto combine = [&](const v8f& c, int ncol) {
      const float bo = bout[(size_t)e * Hq + ncol + lm];
#pragma unroll
      for (int r = 0; r < 8; ++r) {
        const int row = r + 8 * hi;
        const float val = (c[r] + bo) * wS[row];   // padded rows scale to 0
        unsafeAtomicAdd(out + (size_t)tokS[row] * Hq + ncol + lm, val);
      }
    };
    combine(c0, n);
    combine(c1, n + 16);
    combine(c2, n + 32);
    combine(c3, n + 48);
  }
}

// ---------------------------------------------------------------------------
extern "C" void kernel_launch(void* const* d_in, const int* in_sizes, int n_in,
                              void* d_out, int out_size, void* d_ws, size_t ws_size,
                              hipStream_t stream) {
  const float* x    = (const float*)d_in[0];
  const float* Wr   = (const float*)d_in[1];
  const float* br   = (const float*)d_in[2];
  const float* Win  = (const float*)d_in[3];
  const float* bin  = (const float*)d_in[4];
  const float* Wout = (const float*)d_in[5];
  const float* bout = (const float*)d_in[6];
  const float* gum  = (const float*)d_in[7];
  float* out = (float*)d_out;

  // Workspace layout: counts[8] | tok[E*B] | wt[E*B]  (~128 KB)
  int*   counts = (int*)d_ws;
  int*   tok    = counts + Eq;
  float* wtl    = (float*)(tok + Eq * Bq);

  moe_init_kernel<<<(Bq * Hq + 255) / 256, 256, 0, stream>>>(out, counts, tok, wtl);
  moe_router_kernel<<<Bq / 256, 256, 0, stream>>>(x, Wr, br, gum, counts, tok, wtl);

  const size_t lds_bytes = (size_t)(16 * XS_PITCH + 16 * ACT_PITCH) * sizeof(float);
  moe_ffn_kernel<<<Eq * (Bq / 16), 256, lds_bytes, stream>>>(
      x, Win, bin, Wout, bout, counts, tok, wtl, out);
}